// PreUnet3Psi_20255065768096
// MI455X (gfx1250) — compile-verified
//
#include <hip/hip_runtime.h>
#include <math.h>

#ifndef M_PI
#define M_PI 3.14159265358979323846
#endif

typedef float v2f __attribute__((ext_vector_type(2)));
typedef float v8f __attribute__((ext_vector_type(8)));

#define HW    65536          // 256*256
#define NT    12
#define EPS   1e-5f
#define N1    3145728.0f     // B*T*HW
#define N2    262144.0f      // B*HW
#define BTHW  3145728
#define BOHW  8388608

// ---- workspace float-offset map ----
#define ACC_MX    0    // 3: sum x_c
#define ACC_SXX   3    // 6: sum x_c*x_c' (00,01,02,11,12,22)
#define ACC_ST    9
#define ACC_STT   10
#define ACC_SZ    16
#define ACC_SZZ   17
#define ACC_SMIN  32   // 32 (then SMIN2 @64, SMAX @96, SMAX2 @128, contiguous)
#define ACC_ESUM  192  // 128 per (b,o)
#define ACC_ESQ   320  // 128
#define ACC_COUNT 448
// derived parameters
#define PRM      512
#define P_PW     (PRM+0)    // ricker weights [5]
#define P_A0     (PRM+8)    // bn0 scale [32]
#define P_C0     (PRM+40)   // bn0 shift [32]
#define P_AZ     (PRM+72)
#define P_CZ     (PRM+73)
#define P_AT     (PRM+74)
#define P_CT     (PRM+75)
#define P_AMIN   (PRM+80)
#define P_CMIN   (PRM+112)
#define P_AMAX   (PRM+144)
#define P_CMAX   (PRM+176)
#define P_AMEAN  (PRM+208)
#define P_CMEAN  (PRM+240)
#define P_IM     (PRM+256)  // instance-norm mean [128]
#define P_IRS    (PRM+384)  // instance-norm rstd [128]
// bulk data
#define DATA     1024
// tt_pre @ DATA            (BTHW)
// z      @ DATA+BTHW       (BTHW)
// hmin   @ DATA+2*BTHW     (BOHW)
// hmax   @ DATA+2*BTHW+BOHW(BOHW)

// fast silu: x * v_rcp_f32(1+exp(-x))  (avoids IEEE div expansion)
__device__ __forceinline__ float silu(float x) {
  return x * __builtin_amdgcn_rcpf(1.f + __expf(-x));
}

__device__ __forceinline__ v8f wmma4(v2f a, v2f b, v8f c) {
  // V_WMMA_F32_16X16X4_F32 : D = A(16x4) * B(4x16) + C(16x16)
  return __builtin_amdgcn_wmma_f32_16x16x4_f32(false, a, false, b, (short)0, c, false, false);
}

// ---------------------------------------------------------------- prep
__global__ void k_prep(float* ws) {
  int t = threadIdx.x;
  if (t < ACC_COUNT) ws[t] = 0.f;
  if (t == 0) {
    double a = 5.0 / 4.0;
    double A = 2.0 / (sqrt(3.0 * a) * pow(M_PI, 0.25));
    for (int j = 0; j < 5; ++j) {
      double tj = (double)j - 2.0;
      ws[P_PW + j] = (float)(A * (1.0 - (tj / a) * (tj / a)) * exp(-tj * tj / (2.0 * a * a)));
    }
  }
}

// ---------------------------------------------------- pass1: x moments + tt_pre
__global__ void __launch_bounds__(256) k_pass1(const float* __restrict__ x,
                                               const float* __restrict__ wtt,
                                               float* __restrict__ ws) {
  const int gid = blockIdx.x * blockDim.x + threadIdx.x;  // (b,hw) column
  const int b = gid >> 16, hw = gid & (HW - 1);
  float pw[5];
#pragma unroll
  for (int j = 0; j < 5; ++j) pw[j] = ws[P_PW + j];
  const float rw[5] = {-0.2f, -0.1f, 0.f, 0.1f, 0.2f};
  const float wt0 = wtt[0], wt1 = wtt[1], wt2 = wtt[2];

  float xv[3][NT];
#pragma unroll
  for (int c = 0; c < 3; ++c)
#pragma unroll
    for (int t = 0; t < NT; ++t)
      xv[c][t] = x[(size_t)((b * 3 + c) * NT + t) * HW + hw];

  float acc[11];
#pragma unroll
  for (int k = 0; k < 11; ++k) acc[k] = 0.f;
  float* ttp_out = ws + DATA;
#pragma unroll
  for (int t = 0; t < NT; ++t) {
    float trend[3];
#pragma unroll
    for (int c = 0; c < 3; ++c) {
      float pk = 0.f, sl = 0.f;
#pragma unroll
      for (int j = 0; j < 5; ++j) {
        int ti = t - 2 + j;
        float v = (ti >= 0 && ti < NT) ? xv[c][ti] : 0.f;
        pk += pw[j] * v; sl += rw[j] * v;
      }
      trend[c] = pk + fabsf(sl);  // relu(s)+relu(-s) == |s|
    }
    float ttp = wt0 * trend[0] + wt1 * trend[1] + wt2 * trend[2];
    ttp_out[(size_t)(b * NT + t) * HW + hw] = ttp;
    acc[9] += ttp; acc[10] += ttp * ttp;
    float x0 = xv[0][t], x1 = xv[1][t], x2 = xv[2][t];
    acc[0] += x0; acc[1] += x1; acc[2] += x2;
    acc[3] += x0 * x0; acc[4] += x0 * x1; acc[5] += x0 * x2;
    acc[6] += x1 * x1; acc[7] += x1 * x2; acc[8] += x2 * x2;
  }
  __shared__ float red[11];
  if (threadIdx.x < 11) red[threadIdx.x] = 0.f;
  __syncthreads();
#pragma unroll
  for (int k = 0; k < 11; ++k) {
    float v = acc[k];
    for (int m = 16; m >= 1; m >>= 1) v += __shfl_xor(v, m, 32);
    if ((threadIdx.x & 31) == 0) atomicAdd(&red[k], v);
  }
  __syncthreads();
  if (threadIdx.x < 11) atomicAdd(&ws[threadIdx.x], red[threadIdx.x]);
}

// --------------------------------------------- finalize A: bn0 (analytic) + tt bn
__global__ void k_finA(const float* __restrict__ w0, const float* __restrict__ g0,
                       const float* __restrict__ b0, const float* __restrict__ gtt,
                       const float* __restrict__ btt, float* __restrict__ ws) {
  int o = threadIdx.x;
  if (o < 32) {
    float w[3] = {w0[o * 3 + 0], w0[o * 3 + 1], w0[o * 3 + 2]};
    float m0 = ws[0] / N1, m1 = ws[1] / N1, m2 = ws[2] / N1;
    float mean = w[0] * m0 + w[1] * m1 + w[2] * m2;
    float e2 = (w[0] * w[0] * ws[3] + 2.f * w[0] * w[1] * ws[4] + 2.f * w[0] * w[2] * ws[5] +
                w[1] * w[1] * ws[6] + 2.f * w[1] * w[2] * ws[7] + w[2] * w[2] * ws[8]) / N1;
    float var = fmaxf(e2 - mean * mean, 0.f);
    float a = g0[o] * rsqrtf(var + EPS);
    ws[P_A0 + o] = a;
    ws[P_C0 + o] = b0[o] - mean * a;
  }
  if (o == 0) {
    float mt = ws[ACC_ST] / N1;
    float vt = fmaxf(ws[ACC_STT] / N1 - mt * mt, 0.f);
    float at = gtt[0] * rsqrtf(vt + EPS);
    ws[P_AT] = at; ws[P_CT] = btt[0] - mt * at;
  }
}

// --------------- pass2: WMMA conv0 + bn + silu, T-min/max, z = wrt.h, stats
__global__ void __launch_bounds__(256) k_pass2(const float* __restrict__ x,
                                               const float* __restrict__ w0,
                                               const float* __restrict__ wrt,
                                               float* __restrict__ ws) {
  const int lane = threadIdx.x & 31, wave = threadIdx.x >> 5;
  const int half = lane >> 4, l16 = lane & 15;
  const int b = blockIdx.x >> 6;
  const int wbase = (blockIdx.x & 63) * 1024 + wave * 128;
  const float halfmask = half ? 0.f : 1.f;  // branch-free zero-pad for K=c row 3

  // A tiles: A[M=o(16), K=c(4, c=3 zero-pad)], f32 16x4 layout:
  // V0: lanes0-15 K=0 | lanes16-31 K=2 ; V1: K=1 | K=3
  v2f A[2];
#pragma unroll
  for (int m = 0; m < 2; ++m) {
    int o = 16 * m + l16;
    A[m].x = w0[o * 3 + (half ? 2 : 0)];
    A[m].y = halfmask * w0[o * 3 + 1];
  }
  // per-lane bn0/wrt coeffs for its 16 owned o's (D layout: o = 16m + j + 8*half)
  float a0r[2][8], c0r[2][8], wr[2][8];
#pragma unroll
  for (int m = 0; m < 2; ++m)
#pragma unroll
    for (int j = 0; j < 8; ++j) {
      int o = 16 * m + j + 8 * half;
      a0r[m][j] = ws[P_A0 + o]; c0r[m][j] = ws[P_C0 + o]; wr[m][j] = wrt[o];
    }

  float* zbuf  = ws + DATA + BTHW;
  float* hminb = ws + DATA + 2 * BTHW;
  float* hmaxb = hminb + BOHW;

  float mS[2][8] = {}, mS2[2][8] = {}, xS[2][8] = {}, xS2[2][8] = {};
  float sz = 0.f, szz = 0.f;

  for (int tile = 0; tile < 8; ++tile) {
    const int p = wbase + tile * 16 + l16;  // pixel (N dim)
    v8f mn[2], mx[2];
#pragma unroll
    for (int m = 0; m < 2; ++m)
#pragma unroll
      for (int j = 0; j < 8; ++j) { mn[m][j] = 3.0e38f; mx[m][j] = -3.0e38f; }
#pragma unroll
    for (int t = 0; t < NT; ++t) {
      v2f Bm;  // B[K=c, N=pixel], same K-half-split as A
      // Unconditional loads (upper half re-reads cachelines already fetched by
      // lower half); zero-pad via multiply instead of a divergent branch.
      Bm.x = x[(size_t)((b * 3 + (half ? 2 : 0)) * NT + t) * HW + p];
      Bm.y = halfmask * x[(size_t)((b * 3 + 1) * NT + t) * HW + p];
      float zp = 0.f;
#pragma unroll
      for (int m = 0; m < 2; ++m) {
        v8f c;
#pragma unroll
        for (int j = 0; j < 8; ++j) c[j] = 0.f;
        v8f d = wmma4(A[m], Bm, c);  // y0[o,pix] for this t
#pragma unroll
        for (int j = 0; j < 8; ++j) {
          float h = silu(a0r[m][j] * d[j] + c0r[m][j]);
          mn[m][j] = fminf(mn[m][j], h);
          mx[m][j] = fmaxf(mx[m][j], h);
          zp += wr[m][j] * h;
        }
      }
      float z = zp + __shfl_xor(zp, 16, 32);  // combine the two o-halves
      if (!half) {
        zbuf[(size_t)(b * NT + t) * HW + p] = z;
        sz += z; szz += z * z;
      }
    }
#pragma unroll
    for (int m = 0; m < 2; ++m)
#pragma unroll
      for (int j = 0; j < 8; ++j) {
        int o = 16 * m + j + 8 * half;
        size_t ad = (size_t)(b * 32 + o) * HW + p;
        float a = mn[m][j], c = mx[m][j];
        hminb[ad] = a; hmaxb[ad] = c;
        mS[m][j] += a; mS2[m][j] += a * a; xS[m][j] += c; xS2[m][j] += c * c;
      }
  }
  __shared__ float red[128];
  __shared__ float redz[2];
  if (threadIdx.x < 128) red[threadIdx.x] = 0.f;
  if (threadIdx.x < 2) redz[threadIdx.x] = 0.f;
  __syncthreads();
#pragma unroll
  for (int m = 0; m < 2; ++m)
#pragma unroll
    for (int j = 0; j < 8; ++j) {
      float a = mS[m][j], s = mS2[m][j], c = xS[m][j], q = xS2[m][j];
      for (int k = 8; k >= 1; k >>= 1) {
        a += __shfl_xor(a, k, 32); s += __shfl_xor(s, k, 32);
        c += __shfl_xor(c, k, 32); q += __shfl_xor(q, k, 32);
      }
      if (l16 == 0) {
        int o = 16 * m + j + 8 * half;
        atomicAdd(&red[o], a);      atomicAdd(&red[32 + o], s);
        atomicAdd(&red[64 + o], c); atomicAdd(&red[96 + o], q);
      }
    }
  {
    float a = sz, q = szz;
    for (int k = 16; k >= 1; k >>= 1) { a += __shfl_xor(a, k, 32); q += __shfl_xor(q, k, 32); }
    if (lane == 0) { atomicAdd(&redz[0], a); atomicAdd(&redz[1], q); }
  }
  __syncthreads();
  if (threadIdx.x < 128) atomicAdd(&ws[ACC_SMIN + threadIdx.x], red[threadIdx.x]);
  if (threadIdx.x < 2)   atomicAdd(&ws[ACC_SZ + threadIdx.x], redz[threadIdx.x]);
}

// ----------------------------------------- finalize B: rt bn + bn2d coeffs
__global__ void k_finB(const float* grt, const float* brt,
                       const float* g_min, const float* b_min,
                       const float* g_max, const float* b_max,
                       const float* g_mean, const float* b_mean, float* ws) {
  int o = threadIdx.x;
  if (o < 32) {
    float mm = ws[ACC_SMIN + o] / N2;
    float vm = fmaxf(ws[ACC_SMIN + 32 + o] / N2 - mm * mm, 0.f);
    float am = g_min[o] * rsqrtf(vm + EPS);
    ws[P_AMIN + o] = am; ws[P_CMIN + o] = b_min[o] - mm * am;
    float mx = ws[ACC_SMIN + 64 + o] / N2;
    float vx = fmaxf(ws[ACC_SMIN + 96 + o] / N2 - mx * mx, 0.f);
    float ax = g_max[o] * rsqrtf(vx + EPS);
    ws[P_AMAX + o] = ax; ws[P_CMAX + o] = b_max[o] - mx * ax;
    float an = g_mean[o] * rsqrtf(vx + EPS);  // mean branch uses max stats (as in source)
    ws[P_AMEAN + o] = an; ws[P_CMEAN + o] = b_mean[o] - mx * an;
  }
  if (o == 0) {
    float mz = ws[ACC_SZ] / N1;
    float vz = fmaxf(ws[ACC_SZZ] / N1 - mz * mz, 0.f);
    float az = grt[0] * rsqrtf(vz + EPS);
    ws[P_AZ] = az; ws[P_CZ] = brt[0] - mz * az;
  }
}

// --- pass3: WMMA time_to_hidden (K=12 as 3x K=4) + channel branches -> enc + IN stats
__global__ void __launch_bounds__(256) k_pass3(const float* __restrict__ w_th,
                                               const float* __restrict__ b_th,
                                               float* __restrict__ ws,
                                               float* __restrict__ out) {
  const int lane = threadIdx.x & 31, wave = threadIdx.x >> 5;
  const int half = lane >> 4, l16 = lane & 15;
  const int b = blockIdx.x >> 6;
  const int wbase = (blockIdx.x & 63) * 1024 + wave * 128;
  const float az = ws[P_AZ], cz = ws[P_CZ], at = ws[P_AT], ct = ws[P_CT];

  __shared__ float cf[224];  // b_th | a_min c_min | a_max c_max | a_mean c_mean
  __shared__ float red2[64]; // esum[32], esq[32] (this block is within one b)
  if (threadIdx.x < 32) {
    cf[threadIdx.x]        = b_th[threadIdx.x];
    cf[32 + threadIdx.x]   = ws[P_AMIN + threadIdx.x];
    cf[64 + threadIdx.x]   = ws[P_CMIN + threadIdx.x];
    cf[96 + threadIdx.x]   = ws[P_AMAX + threadIdx.x];
    cf[128 + threadIdx.x]  = ws[P_CMAX + threadIdx.x];
    cf[160 + threadIdx.x]  = ws[P_AMEAN + threadIdx.x];
    cf[192 + threadIdx.x]  = ws[P_CMEAN + threadIdx.x];
  }
  if (threadIdx.x < 64) red2[threadIdx.x] = 0.f;
  __syncthreads();

  // A tiles for w_th: A[M=o, K=t], 3 chained K=4 steps, 2 o-tiles
  v2f A[2][3];
#pragma unroll
  for (int m = 0; m < 2; ++m)
#pragma unroll
    for (int g = 0; g < 3; ++g) {
      int o = 16 * m + l16;
      A[m][g].x = w_th[o * NT + 4 * g + (half ? 2 : 0)];
      A[m][g].y = w_th[o * NT + 4 * g + (half ? 3 : 1)];
    }

  const float* ttp   = ws + DATA;
  const float* zbuf  = ws + DATA + BTHW;
  const float* hminb = ws + DATA + 2 * BTHW;
  const float* hmaxb = hminb + BOHW;

  float eS[2][8] = {}, eQ[2][8] = {};

  for (int tile = 0; tile < 8; ++tile) {
    const int p = wbase + tile * 16 + l16;
    v2f Bm[3];  // s = rt + tt, B[K=t,N=pixel]
#pragma unroll
    for (int g = 0; g < 3; ++g) {
      int t0 = 4 * g + (half ? 2 : 0), t1 = 4 * g + (half ? 3 : 1);
      size_t i0 = (size_t)(b * NT + t0) * HW + p, i1 = (size_t)(b * NT + t1) * HW + p;
      Bm[g].x = silu(az * zbuf[i0] + cz) + silu(at * ttp[i0] + ct);
      Bm[g].y = silu(az * zbuf[i1] + cz) + silu(at * ttp[i1] + ct);
    }
#pragma unroll
    for (int m = 0; m < 2; ++m) {
      v8f c;
#pragma unroll
      for (int j = 0; j < 8; ++j) c[j] = cf[16 * m + j + 8 * half];  // seed with b_th
      c = wmma4(A[m][0], Bm[0], c);
      c = wmma4(A[m][1], Bm[1], c);
      c = wmma4(A[m][2], Bm[2], c);
#pragma unroll
      for (int j = 0; j < 8; ++j) {
        int o = 16 * m + j + 8 * half;
        size_t ad = (size_t)(b * 32 + o) * HW + p;
        float hmn = hminb[ad], hmx = hmaxb[ad];
        float enc = c[j] + silu(cf[32 + o] * hmn + cf[64 + o])
                         + silu(cf[96 + o] * hmx + cf[128 + o])
                         + silu(cf[160 + o] * hmx + cf[192 + o]);
        out[ad] = enc;
        eS[m][j] += enc; eQ[m][j] += enc * enc;
      }
    }
  }
#pragma unroll
  for (int m = 0; m < 2; ++m)
#pragma unroll
    for (int j = 0; j < 8; ++j) {
      float a = eS[m][j], q = eQ[m][j];
      for (int k = 8; k >= 1; k >>= 1) { a += __shfl_xor(a, k, 32); q += __shfl_xor(q, k, 32); }
      if (l16 == 0) {
        int o = 16 * m + j + 8 * half;
        atomicAdd(&red2[o], a); atomicAdd(&red2[32 + o], q);
      }
    }
  __syncthreads();
  if (threadIdx.x < 32)       atomicAdd(&ws[ACC_ESUM + b * 32 + threadIdx.x], red2[threadIdx.x]);
  else if (threadIdx.x < 64)  atomicAdd(&ws[ACC_ESQ + b * 32 + threadIdx.x - 32], red2[threadIdx.x]);
}

// ------------------------------------------------ finalize C: instance-norm stats
__global__ void k_finC(float* ws) {
  int i = threadIdx.x;
  if (i < 128) {
    float m = ws[ACC_ESUM + i] / 65536.0f;
    float v = fmaxf(ws[ACC_ESQ + i] / 65536.0f - m * m, 0.f);
    ws[P_IM + i] = m; ws[P_IRS + i] = rsqrtf(v + EPS);
  }
}

// ------------------------------------------------------- pass4: normalize in place
__global__ void __launch_bounds__(256) k_pass4(float* __restrict__ out,
                                               const float* __restrict__ ws) {
  size_t base = ((size_t)blockIdx.x * blockDim.x + threadIdx.x) * 4;
  int bo = (int)(base >> 16);
  float m = ws[P_IM + bo], r = ws[P_IRS + bo];
  float4 v = *reinterpret_cast<float4*>(out + base);
  v.x = (v.x - m) * r; v.y = (v.y - m) * r; v.z = (v.z - m) * r; v.w = (v.w - m) * r;
  *reinterpret_cast<float4*>(out + base) = v;
}

extern "C" void kernel_launch(void* const* d_in, const int* in_sizes, int n_in,
                              void* d_out, int out_size, void* d_ws, size_t ws_size,
                              hipStream_t stream) {
  const float* x      = (const float*)d_in[0];
  const float* w0     = (const float*)d_in[1];
  const float* g0     = (const float*)d_in[2];
  const float* b0     = (const float*)d_in[3];
  const float* wrt    = (const float*)d_in[4];
  const float* grt    = (const float*)d_in[5];
  const float* brt    = (const float*)d_in[6];
  const float* wtt    = (const float*)d_in[7];
  const float* gtt    = (const float*)d_in[8];
  const float* btt    = (const float*)d_in[9];
  const float* w_th   = (const float*)d_in[10];
  const float* b_th   = (const float*)d_in[11];
  const float* g_min  = (const float*)d_in[12];
  const float* b_min  = (const float*)d_in[13];
  const float* g_max  = (const float*)d_in[14];
  const float* b_max  = (const float*)d_in[15];
  const float* g_mean = (const float*)d_in[16];
  const float* b_mean = (const float*)d_in[17];
  float* ws  = (float*)d_ws;   // needs ~92.3 MB
  float* out = (float*)d_out;  // (4,32,256,256)

  k_prep <<<1, 512, 0, stream>>>(ws);
  k_pass1<<<1024, 256, 0, stream>>>(x, wtt, ws);
  k_finA <<<1, 32, 0, stream>>>(w0, g0, b0, gtt, btt, ws);
  k_pass2<<<256, 256, 0, stream>>>(x, w0, wrt, ws);
  k_finB <<<1, 32, 0, stream>>>(grt, brt, g_min, b_min, g_max, b_max, g_mean, b_mean, ws);
  k_pass3<<<256, 256, 0, stream>>>(w_th, b_th, ws, out);
  k_finC <<<1, 128, 0, stream>>>(ws);
  k_pass4<<<2097152 / 256, 256, 0, stream>>>(out, ws);
}